// SoftHGNN_72851235274890
// MI455X (gfx1250) — compile-verified
//
#include <hip/hip_runtime.h>
#include <math.h>

#define BB 8
#define NN 8192
#define DD 256
#define E1 8
#define E2 4
#define NE 16
#define NCHUNK 32
#define RPC (NN / NCHUNK) /* 256 rows per chunk */

typedef __attribute__((ext_vector_type(2))) float v2f;
typedef __attribute__((ext_vector_type(8))) float v8f;

__device__ __forceinline__ float gelu_exact(float x) {
  return 0.5f * x * (1.0f + erff(x * 0.70710678118654752440f));
}

// ---------- ctx = [mean_n, max_n] over the big X ----------
__global__ void k_reduce_partial(const float* __restrict__ X,
                                 float* __restrict__ psum,
                                 float* __restrict__ pmax) {
  int b = blockIdx.x, c = blockIdx.y, d = threadIdx.x;
  const float* p = X + ((size_t)b * NN + (size_t)c * RPC) * DD + d;
  float s = 0.f, m = -3.402823466e38f;
  for (int r = 0; r < RPC; ++r) {
    float v = p[(size_t)r * DD];
    s += v;
    m = fmaxf(m, v);
  }
  psum[(b * NCHUNK + c) * DD + d] = s;
  pmax[(b * NCHUNK + c) * DD + d] = m;
}

__global__ void k_reduce_final(const float* __restrict__ psum,
                               const float* __restrict__ pmax,
                               float* __restrict__ ctx) {
  int b = blockIdx.x, d = threadIdx.x;
  float s = 0.f, m = -3.402823466e38f;
  for (int c = 0; c < NCHUNK; ++c) {
    s += psum[(b * NCHUNK + c) * DD + d];
    m = fmaxf(m, pmax[(b * NCHUNK + c) * DD + d]);
  }
  ctx[b * (2 * DD) + d] = s / (float)NN;
  ctx[b * (2 * DD) + DD + d] = m;
}

// ---------- P[b,e,:] = proto[e,:] + ctx[b,:] @ cwT + cb ----------
__global__ void k_proto(const float* __restrict__ ctx, const float* __restrict__ cw,
                        const float* __restrict__ cb, const float* __restrict__ proto,
                        float* __restrict__ P, int E) {
  __shared__ float cs[2 * DD];
  int b = blockIdx.x, e = blockIdx.y, t = threadIdx.x;
  cs[t] = ctx[b * (2 * DD) + t];
  cs[DD + t] = ctx[b * (2 * DD) + DD + t];
  __syncthreads();
  int r = e * DD + t;
  const float* wr = cw + (size_t)r * (2 * DD);
  float acc = cb[r];
  for (int k = 0; k < 2 * DD; ++k) acc += wr[k] * cs[k];
  P[((size_t)b * E + e) * DD + t] = proto[r] + acc;
}

// ---------- W2[b,d,e] = (1/32) * sum_k pw[k,d]*P[b,e,k] ----------
__global__ void k_w2(const float* __restrict__ P, const float* __restrict__ pw,
                     float* __restrict__ W2) {
  __shared__ float Ps[E1][DD];
  int b = blockIdx.x, d = threadIdx.x;
  for (int j = d; j < E1 * DD; j += 256) Ps[j >> 8][j & 255] = P[(size_t)b * E1 * DD + j];
  __syncthreads();
  float acc[E1];
#pragma unroll
  for (int e = 0; e < E1; ++e) acc[e] = 0.f;
  for (int k = 0; k < DD; ++k) {
    float w = pw[k * DD + d];
#pragma unroll
    for (int e = 0; e < E1; ++e) acc[e] += w * Ps[e][k];
  }
#pragma unroll
  for (int e = 0; e < E1; ++e)
    W2[((size_t)b * DD + d) * E1 + e] = acc[e] * (1.0f / 32.0f);
}

// ---------- logits = X @ W2 via V_WMMA_F32_16X16X4_F32 ----------
__global__ void k_logits_wmma(const float* __restrict__ X, const float* __restrict__ W2,
                              float* __restrict__ L) {
  __shared__ float Ws[DD][16]; // K x (E padded to 16); row stride 16 -> +2 rows = +32 banks
  int b = blockIdx.x, t = threadIdx.x;
  for (int j = t; j < DD * 16; j += 256) {
    int row = j >> 4, col = j & 15;
    Ws[row][col] = (col < E1) ? W2[((size_t)b * DD + row) * E1 + col] : 0.f;
  }
  __syncthreads();
  int wave = t >> 5, lane = t & 31;
  int m15 = lane & 15, hi = lane >> 4, kb = hi * 2;
  int rowBase = blockIdx.y * 128 + wave * 16;
  const float* xr = X + ((size_t)b * NN + rowBase + m15) * DD;
  v8f c = {};
  for (int kk = 0; kk < DD / 4; ++kk) {
    int k0 = kk * 4;
    v2f a = *(const v2f*)(xr + k0 + kb);
    v2f bm;
    bm[0] = Ws[k0 + kb][m15];
    bm[1] = Ws[k0 + kb + 1][m15];
    c = __builtin_amdgcn_wmma_f32_16x16x4_f32(false, a, false, bm, (short)0, c, false, false);
  }
  if (m15 < E1) {
    int rb = hi * 8;
#pragma unroll
    for (int v = 0; v < 8; ++v)
      L[((size_t)b * NN + rowBase + rb + v) * E1 + m15] = c[v];
  }
}

// ---------- softmax over the node axis (N) per (b,e) ----------
__global__ void k_softmax_stats(const float* __restrict__ L, float* __restrict__ stat) {
  __shared__ float red[256];
  int b = blockIdx.x, e = blockIdx.y, t = threadIdx.x;
  const float* base = L + (size_t)b * NN * E1 + e;
  float m = -3.402823466e38f;
  for (int n = t; n < NN; n += 256) m = fmaxf(m, base[(size_t)n * E1]);
  red[t] = m;
  __syncthreads();
  for (int q = 128; q > 0; q >>= 1) {
    if (t < q) red[t] = fmaxf(red[t], red[t + q]);
    __syncthreads();
  }
  m = red[0];
  __syncthreads();
  float s = 0.f;
  for (int n = t; n < NN; n += 256) s += expf(base[(size_t)n * E1] - m);
  red[t] = s;
  __syncthreads();
  for (int q = 128; q > 0; q >>= 1) {
    if (t < q) red[t] += red[t + q];
    __syncthreads();
  }
  if (t == 0) {
    stat[(b * E1 + e) * 2] = m;
    stat[(b * E1 + e) * 2 + 1] = red[0];
  }
}

__global__ void k_softmax_apply(float* __restrict__ A, const float* __restrict__ stat) {
  size_t i = (size_t)blockIdx.x * 256 + threadIdx.x;
  int e = (int)(i & 7);
  int b = (int)(i / ((size_t)NN * E1));
  float m = stat[(b * E1 + e) * 2], s = stat[(b * E1 + e) * 2 + 1];
  A[i] = expf(A[i] - m) / s;
}

// ---------- S[b,e,:] = A^T X  (deterministic 2-stage) ----------
__global__ void k_edge_accum(const float* __restrict__ X, const float* __restrict__ A,
                             float* __restrict__ Spart) {
  __shared__ float As[RPC][E1];
  int b = blockIdx.x, c = blockIdx.y, d = threadIdx.x;
  const float* ab = A + ((size_t)b * NN + (size_t)c * RPC) * E1;
  for (int j = d; j < RPC * E1; j += 256) As[j >> 3][j & 7] = ab[j];
  __syncthreads();
  const float* xb = X + ((size_t)b * NN + (size_t)c * RPC) * DD + d;
  float acc[E1];
#pragma unroll
  for (int e = 0; e < E1; ++e) acc[e] = 0.f;
  for (int r = 0; r < RPC; ++r) {
    float x = xb[(size_t)r * DD];
#pragma unroll
    for (int e = 0; e < E1; ++e) acc[e] += As[r][e] * x;
  }
#pragma unroll
  for (int e = 0; e < E1; ++e)
    Spart[(((size_t)b * NCHUNK + c) * E1 + e) * DD + d] = acc[e];
}

__global__ void k_edge_finalize(const float* __restrict__ Spart, const float* __restrict__ w,
                                const float* __restrict__ bias, float* __restrict__ He,
                                int rowOff) {
  __shared__ float srow[DD];
  int b = blockIdx.x, e = blockIdx.y, d = threadIdx.x;
  float s = 0.f;
  for (int c = 0; c < NCHUNK; ++c)
    s += Spart[(((size_t)b * NCHUNK + c) * E1 + e) * DD + d];
  srow[d] = s;
  __syncthreads();
  const float* wr = w + (size_t)d * DD;
  float acc = bias[d];
  for (int k = 0; k < DD; ++k) acc += wr[k] * srow[k];
  He[((size_t)b * NE + rowOff + e) * DD + d] = gelu_exact(acc);
}

// ---------- cross-modal (tiny: 16 nodes) ----------
__global__ void k_cm_ctx(const float* __restrict__ He, float* __restrict__ ctx) {
  int b = blockIdx.x, d = threadIdx.x;
  float s = 0.f, m = -3.402823466e38f;
  for (int n = 0; n < NE; ++n) {
    float v = He[((size_t)b * NE + n) * DD + d];
    s += v;
    m = fmaxf(m, v);
  }
  ctx[b * (2 * DD) + d] = s / (float)NE;
  ctx[b * (2 * DD) + DD + d] = m;
}

__global__ void k_cm_logits(const float* __restrict__ He, const float* __restrict__ pw,
                            const float* __restrict__ pb, const float* __restrict__ P,
                            float* __restrict__ Lcm) {
  __shared__ float row[DD], xp[DD], red[DD];
  int b = blockIdx.x, n = blockIdx.y, t = threadIdx.x;
  row[t] = He[((size_t)b * NE + n) * DD + t];
  __syncthreads();
  const float* wr = pw + (size_t)t * DD;
  float acc = pb[t];
  for (int k = 0; k < DD; ++k) acc += wr[k] * row[k];
  xp[t] = acc;
  __syncthreads();
  for (int e = 0; e < E2; ++e) {
    red[t] = xp[t] * P[((size_t)b * E2 + e) * DD + t];
    __syncthreads();
    for (int q = 128; q > 0; q >>= 1) {
      if (t < q) red[t] += red[t + q];
      __syncthreads();
    }
    if (t == 0) Lcm[(b * NE + n) * E2 + e] = red[0] * (1.0f / 32.0f);
    __syncthreads();
  }
}

__global__ void k_cm_softmax(const float* __restrict__ Lcm, float* __restrict__ Acm) {
  int b = blockIdx.x, t = threadIdx.x;
  if (t < E2) {
    float m = -3.402823466e38f;
    for (int n = 0; n < NE; ++n) m = fmaxf(m, Lcm[(b * NE + n) * E2 + t]);
    float s = 0.f;
    for (int n = 0; n < NE; ++n) s += expf(Lcm[(b * NE + n) * E2 + t] - m);
    for (int n = 0; n < NE; ++n)
      Acm[(b * NE + n) * E2 + t] = expf(Lcm[(b * NE + n) * E2 + t] - m) / s;
  }
}

__global__ void k_cm_scm(const float* __restrict__ He, const float* __restrict__ Acm,
                         float* __restrict__ Scm) {
  int b = blockIdx.x, e = blockIdx.y, d = threadIdx.x;
  float s = 0.f;
  for (int n = 0; n < NE; ++n)
    s += Acm[(b * NE + n) * E2 + e] * He[((size_t)b * NE + n) * DD + d];
  Scm[((size_t)b * E2 + e) * DD + d] = s;
}

__global__ void k_row_linear_gelu(const float* __restrict__ in, const float* __restrict__ w,
                                  const float* __restrict__ bias, float* __restrict__ out,
                                  int rows) {
  __shared__ float srow[DD];
  int b = blockIdx.x, r = blockIdx.y, t = threadIdx.x;
  size_t idx = ((size_t)b * rows + r) * DD;
  srow[t] = in[idx + t];
  __syncthreads();
  const float* wr = w + (size_t)t * DD;
  float acc = bias[t];
  for (int k = 0; k < DD; ++k) acc += wr[k] * srow[k];
  out[idx + t] = gelu_exact(acc);
}

__global__ void k_cm_update(const float* __restrict__ Acm, const float* __restrict__ Hcm,
                            const float* __restrict__ w, const float* __restrict__ bias,
                            float* __restrict__ He) {
  __shared__ float Ts[DD];
  __shared__ float a[E2];
  int b = blockIdx.x, n = blockIdx.y, t = threadIdx.x;
  if (t < E2) a[t] = Acm[(b * NE + n) * E2 + t];
  __syncthreads();
  float s = 0.f;
#pragma unroll
  for (int e = 0; e < E2; ++e) s += a[e] * Hcm[((size_t)b * E2 + e) * DD + t];
  Ts[t] = s;
  __syncthreads();
  const float* wr = w + (size_t)t * DD;
  float acc = bias[t];
  for (int k = 0; k < DD; ++k) acc += wr[k] * Ts[k];
  float val = gelu_exact(acc);
  float scale = (n < E1) ? 1.0f : 3.0f; // He_ir gets 3x the add (source bug kept)
  He[((size_t)b * NE + n) * DD + t] += scale * val;
}

// ---------- M[b,e,:] = He_final[b,e,:] @ np_w^T ----------
__global__ void k_make_M(const float* __restrict__ He, const float* __restrict__ w,
                         float* __restrict__ M, int rowOff) {
  __shared__ float row[DD];
  int b = blockIdx.x, e = blockIdx.y, d = threadIdx.x;
  size_t idx = ((size_t)b * NE + rowOff + e) * DD;
  row[d] = He[idx + d];
  __syncthreads();
  const float* wr = w + (size_t)d * DD;
  float acc = 0.f;
  for (int k = 0; k < DD; ++k) acc += wr[k] * row[k];
  M[idx + d] = acc;
}

// ---------- out = gelu(A@M + b) + X, via V_WMMA_F32_16X16X4_F32 ----------
__global__ void k_final_wmma(const float* __restrict__ X, const float* __restrict__ A,
                             const float* __restrict__ M, const float* __restrict__ bias,
                             float* __restrict__ out, int colOff, int mOff) {
  __shared__ float Ms[E1][DD + 16]; // stride 272 -> rows +2 shift banks by 32 (conflict-free)
  __shared__ float bs[DD];
  int b = blockIdx.x, t = threadIdx.x;
  for (int j = t; j < E1 * DD; j += 256)
    Ms[j >> 8][j & 255] = M[((size_t)b * NE + mOff) * DD + j];
  bs[t] = bias[t];
  __syncthreads();
  int wave = t >> 5, lane = t & 31;
  int m15 = lane & 15, hi = lane >> 4, kb = hi * 2;
  int rowBase = blockIdx.y * 128 + wave * 16;
  const float* ar = A + ((size_t)b * NN + rowBase + m15) * E1;
  v2f a0 = *(const v2f*)(ar + kb);
  v2f a1 = *(const v2f*)(ar + 4 + kb);
  int rb = hi * 8;
  for (int dt = 0; dt < DD / 16; ++dt) {
    int dcol = dt * 16 + m15;
    v2f b0, b1;
    b0[0] = Ms[kb][dcol];
    b0[1] = Ms[kb + 1][dcol];
    b1[0] = Ms[4 + kb][dcol];
    b1[1] = Ms[5 + kb][dcol];
    v8f c = {};
    c = __builtin_amdgcn_wmma_f32_16x16x4_f32(false, a0, false, b0, (short)0, c, false, false);
    c = __builtin_amdgcn_wmma_f32_16x16x4_f32(false, a1, false, b1, (short)0, c, false, false);
#pragma unroll
    for (int v = 0; v < 8; ++v) {
      size_t n = (size_t)b * NN + rowBase + rb + v;
      float val = gelu_exact(c[v] + bs[dcol]) + X[n * DD + dcol];
      out[n * (2 * DD) + colOff + dcol] = val;
    }
  }
}

extern "C" void kernel_launch(void* const* d_in, const int* in_sizes, int n_in,
                              void* d_out, int out_size, void* d_ws, size_t ws_size,
                              hipStream_t stream) {
  (void)in_sizes; (void)n_in; (void)out_size; (void)ws_size;
  const float* x_mod[2] = {(const float*)d_in[0], (const float*)d_in[1]};
  const float* g_cm_proto = (const float*)d_in[12];
  const float* g_cm_cw = (const float*)d_in[13];
  const float* g_cm_cb = (const float*)d_in[14];
  const float* g_cm_pw = (const float*)d_in[15];
  const float* g_cm_pb = (const float*)d_in[16];
  const float* ep_cm_w = (const float*)d_in[25];
  const float* ep_cm_b = (const float*)d_in[26];
  const float* np_cm_w = (const float*)d_in[27];
  const float* np_cm_b = (const float*)d_in[28];

  float* w = (float*)d_ws;
  size_t o = 0;
  float* A_mod[2];
  A_mod[0] = w + o; o += (size_t)BB * NN * E1;
  A_mod[1] = w + o; o += (size_t)BB * NN * E1;
  float* psum = w + o; o += (size_t)BB * NCHUNK * DD;
  float* pmax = w + o; o += (size_t)BB * NCHUNK * DD;
  float* ctxb = w + o; o += (size_t)BB * 2 * DD;
  float* Pbuf = w + o; o += (size_t)BB * E1 * DD;
  float* W2buf = w + o; o += (size_t)BB * DD * E1;
  float* statb = w + o; o += (size_t)BB * E1 * 2;
  float* Spart = w + o; o += (size_t)BB * NCHUNK * E1 * DD;
  float* HeF = w + o; o += (size_t)BB * NE * DD;
  float* ctxcm = w + o; o += (size_t)BB * 2 * DD;
  float* Pcm = w + o; o += (size_t)BB * E2 * DD;
  float* Lcm = w + o; o += (size_t)BB * NE * E2;
  float* Acm = w + o; o += (size_t)BB * NE * E2;
  float* Scm = w + o; o += (size_t)BB * E2 * DD;
  float* Hcm = w + o; o += (size_t)BB * E2 * DD;
  float* Mbuf = w + o; o += (size_t)BB * NE * DD;

  // per-modality: hyperedge_gen (fused) + He_m = gelu(linear(A^T X))
  for (int m = 0; m < 2; ++m) {
    const float* X = x_mod[m];
    float* A = A_mod[m];
    const float* proto = (const float*)d_in[2 + 5 * m];
    const float* cw = (const float*)d_in[3 + 5 * m];
    const float* cb = (const float*)d_in[4 + 5 * m];
    const float* pw = (const float*)d_in[5 + 5 * m];
    const float* ep_w = (const float*)d_in[17 + 4 * m];
    const float* ep_b = (const float*)d_in[18 + 4 * m];

    k_reduce_partial<<<dim3(BB, NCHUNK), 256, 0, stream>>>(X, psum, pmax);
    k_reduce_final<<<BB, 256, 0, stream>>>(psum, pmax, ctxb);
    k_proto<<<dim3(BB, E1), 256, 0, stream>>>(ctxb, cw, cb, proto, Pbuf, E1);
    k_w2<<<BB, 256, 0, stream>>>(Pbuf, pw, W2buf);
    k_logits_wmma<<<dim3(BB, NN / 128), 256, 0, stream>>>(X, W2buf, A);
    k_softmax_stats<<<dim3(BB, E1), 256, 0, stream>>>(A, statb);
    k_softmax_apply<<<(BB * NN * E1) / 256, 256, 0, stream>>>(A, statb);
    k_edge_accum<<<dim3(BB, NCHUNK), 256, 0, stream>>>(X, A, Spart);
    k_edge_finalize<<<dim3(BB, E1), 256, 0, stream>>>(Spart, ep_w, ep_b, HeF, m * E1);
  }

  // cross-modal hyperedges over the 16 fused edges (all tiny)
  k_cm_ctx<<<BB, 256, 0, stream>>>(HeF, ctxcm);
  k_proto<<<dim3(BB, E2), 256, 0, stream>>>(ctxcm, g_cm_cw, g_cm_cb, g_cm_proto, Pcm, E2);
  k_cm_logits<<<dim3(BB, NE), 256, 0, stream>>>(HeF, g_cm_pw, g_cm_pb, Pcm, Lcm);
  k_cm_softmax<<<BB, 32, 0, stream>>>(Lcm, Acm);
  k_cm_scm<<<dim3(BB, E2), 256, 0, stream>>>(HeF, Acm, Scm);
  k_row_linear_gelu<<<dim3(BB, E2), 256, 0, stream>>>(Scm, ep_cm_w, ep_cm_b, Hcm, E2);
  k_cm_update<<<dim3(BB, NE), 256, 0, stream>>>(Acm, Hcm, np_cm_w, np_cm_b, HeF);

  // final node projection fused with gelu + residual + concat store
  k_make_M<<<dim3(BB, E1), 256, 0, stream>>>(HeF, (const float*)d_in[19], Mbuf, 0);
  k_make_M<<<dim3(BB, E1), 256, 0, stream>>>(HeF, (const float*)d_in[23], Mbuf, E1);
  float* out = (float*)d_out;
  k_final_wmma<<<dim3(BB, NN / 128), 256, 0, stream>>>(
      x_mod[0], A_mod[0], Mbuf, (const float*)d_in[20], out, 0, 0);
  k_final_wmma<<<dim3(BB, NN / 128), 256, 0, stream>>>(
      x_mod[1], A_mod[1], Mbuf, (const float*)d_in[24], out, DD, E1);
}